// MultiTaskHead_49667001811100
// MI455X (gfx1250) — compile-verified
//
#include <hip/hip_runtime.h>
#include <hip/hip_bf16.h>
#include <math.h>

// ---------------------------------------------------------------------------
// CDNA5 (gfx1250) fp32 WMMA: V_WMMA_F32_16X16X4_F32, wave32.
// A (16x4): lanes 0-15 rows M=0..15, K={k0,k0+1}; lanes 16-31 K={k0+2,k0+3}.
// B (4x16): VGPR i holds rows K=i (lanes 0-15) / K=i+2 (lanes 16-31).
// C/D: VGPR r = row r (lanes 0-15) / row r+8 (lanes 16-31), col = lane&15.
//
// Weights are pre-swizzled in PAIRED fragment order: for each k-step, the B
// fragments of two adjacent n-tiles are interleaved per lane, so ONE
// global_load_b128 per lane feeds 2 n-tiles. With 2-way M blocking each
// inner step issues 1 b128 + 2 ds_b64 for 4 WMMAs (0.75 mem ops / wmma).
// ---------------------------------------------------------------------------
typedef __attribute__((ext_vector_type(2))) float v2f;
typedef __attribute__((ext_vector_type(4))) float v4f;
typedef __attribute__((ext_vector_type(8))) float v8f;

__device__ __forceinline__ void lds_fence() {
  asm volatile("s_wait_dscnt 0" ::: "memory");
}

__device__ __forceinline__ v8f wmma4(v2f a, v2f b, v8f c) {
  return __builtin_amdgcn_wmma_f32_16x16x4_f32(false, a, false, b, (short)0, c,
                                               false, false);
}

// A fragment from LDS tile (even stride -> 8B aligned -> ds_load_b64)
__device__ __forceinline__ v2f lda(const float* X, int sx, int kt, int lane, int rofs) {
  int row = (lane & 15) + rofs;
  int kk  = kt * 4 + ((lane >> 4) << 1);
  return *(const v2f*)(&X[row * sx + kk]);
}

// 2x2 (M x N) register-blocked GEMM: rows 0-15/16-31 of X, n-tile pair fp.
__device__ __forceinline__ void gemm_22(const float* X, int sx, const float* Wsw,
                                        int nPairs, int K4, int fp, int lane,
                                        v8f& d00, v8f& d10, v8f& d01, v8f& d11) {
  v8f a00 = {}, a10 = {}, a01 = {}, a11 = {};
  for (int kt = 0; kt < K4; ++kt) {
    v4f t = *(const v4f*)(&Wsw[(((size_t)kt * nPairs) + fp) * 128 + lane * 4]);
    v2f b0; b0.x = t.x; b0.y = t.y;
    v2f b1; b1.x = t.z; b1.y = t.w;
    v2f x0 = lda(X, sx, kt, lane, 0);
    v2f x1 = lda(X, sx, kt, lane, 16);
    a00 = wmma4(x0, b0, a00);
    a10 = wmma4(x1, b0, a10);
    a01 = wmma4(x0, b1, a01);
    a11 = wmma4(x1, b1, a11);
  }
  d00 = a00; d10 = a10; d01 = a01; d11 = a11;
}

__device__ __forceinline__ float silu(float x) { return x / (1.f + __expf(-x)); }

#define NNODE 50000
#define NEDGE 500000
#define NTGT  10000
#define ETGT  100000

// ---------------------------------------------------------------------------
// Weight swizzle into paired fragment order, zero-padded to Kp x Np (Np%32==0).
// chunk (kt, fp) = 128 floats; lane owns [b0.x b0.y b1.x b1.y].
// ---------------------------------------------------------------------------
__global__ void swizzle_w(const float* __restrict__ W, int Kw, int Nw,
                          int Kp, int Np, float* __restrict__ out) {
  int i = blockIdx.x * blockDim.x + threadIdx.x;   // (kt, fp, lane)
  int total = (Kp >> 2) * (Np >> 5) * 32;
  if (i >= total) return;
  int lane   = i & 31;
  int f      = i >> 5;
  int nPairs = Np >> 5;
  int kt = f / nPairs, fp = f - kt * nPairs;
  int c0 = fp * 32 + (lane & 15);
  int c1 = c0 + 16;
  int kk = kt * 4 + ((lane >> 4) << 1);
  float4 o;
  o.x = (kk     < Kw && c0 < Nw) ? W[(size_t)kk * Nw + c0]       : 0.f;
  o.y = (kk + 1 < Kw && c0 < Nw) ? W[(size_t)(kk + 1) * Nw + c0] : 0.f;
  o.z = (kk     < Kw && c1 < Nw) ? W[(size_t)kk * Nw + c1]       : 0.f;
  o.w = (kk + 1 < Kw && c1 < Nw) ? W[(size_t)(kk + 1) * Nw + c1] : 0.f;
  *(float4*)(&out[(size_t)i * 4]) = o;
}

__global__ void zero_f32(float* __restrict__ p, int n) {
  int i = blockIdx.x * blockDim.x + threadIdx.x;
  if (i < n) p[i] = 0.f;
}

// ---------------------------------------------------------------------------
// Kernel 1: per-node precompute (32 nodes per wave).
// ---------------------------------------------------------------------------
__global__ void __launch_bounds__(32) node_precompute(
    const float* __restrict__ h,
    const float* __restrict__ sw_nsp,  const float* __restrict__ b_nsp,
    const float* __restrict__ sw_left, const float* __restrict__ b_left,
    const float* __restrict__ sw_right,const float* __restrict__ b_right,
    const float* __restrict__ sw_tp0,  const float* __restrict__ sw_tp1,
    float* __restrict__ hs, float* __restrict__ lft, float* __restrict__ rgt,
    float* __restrict__ atp, float* __restrict__ pv)
{
  __shared__ float Xs[32 * 132];
  __shared__ float Bs[32 * 132];
  const int lane = threadIdx.x;
  const int n0r  = blockIdx.x * 32;

  for (int i = lane; i < 32 * 32; i += 32) {
    int r = i >> 5, c = (i & 31) << 2;
    int rr = n0r + r; if (rr > NNODE - 1) rr = NNODE - 1;
    *(float4*)(&Xs[r * 132 + c]) = *(const float4*)(h + (size_t)rr * 320 + c);
  }
  lds_fence();

  // hs = X @ W_nsp + b
  for (int fp = 0; fp < 4; ++fp) {
    v8f d00, d10, d01, d11;
    gemm_22(Xs, 132, sw_nsp, 4, 32, fp, lane, d00, d10, d01, d11);
    int c0 = fp * 32 + (lane & 15), c1 = c0 + 16;
    float bb0 = b_nsp[c0], bb1 = b_nsp[c1];
    for (int r = 0; r < 8; ++r) {
      int row = r + ((lane >> 4) << 3);
      float v00 = d00[r] + bb0, v10 = d10[r] + bb0;
      float v01 = d01[r] + bb1, v11 = d11[r] + bb1;
      Bs[row * 132 + c0] = v00;         Bs[(row + 16) * 132 + c0] = v10;
      Bs[row * 132 + c1] = v01;         Bs[(row + 16) * 132 + c1] = v11;
      if (n0r + row < NNODE) {
        hs[(size_t)(n0r + row) * 128 + c0] = v00;
        hs[(size_t)(n0r + row) * 128 + c1] = v01;
      }
      if (n0r + row + 16 < NNODE) {
        hs[(size_t)(n0r + row + 16) * 128 + c0] = v10;
        hs[(size_t)(n0r + row + 16) * 128 + c1] = v11;
      }
    }
  }
  lds_fence();

  // left / right = hs @ W + b
  for (int fp = 0; fp < 4; ++fp) {
    v8f d00, d10, d01, d11;
    gemm_22(Bs, 132, sw_left, 4, 32, fp, lane, d00, d10, d01, d11);
    int c0 = fp * 32 + (lane & 15), c1 = c0 + 16;
    float bb0 = b_left[c0], bb1 = b_left[c1];
    for (int r = 0; r < 8; ++r) {
      int row = r + ((lane >> 4) << 3);
      if (n0r + row < NNODE) {
        lft[(size_t)(n0r + row) * 128 + c0] = d00[r] + bb0;
        lft[(size_t)(n0r + row) * 128 + c1] = d01[r] + bb1;
      }
      if (n0r + row + 16 < NNODE) {
        lft[(size_t)(n0r + row + 16) * 128 + c0] = d10[r] + bb0;
        lft[(size_t)(n0r + row + 16) * 128 + c1] = d11[r] + bb1;
      }
    }
  }
  for (int fp = 0; fp < 4; ++fp) {
    v8f d00, d10, d01, d11;
    gemm_22(Bs, 132, sw_right, 4, 32, fp, lane, d00, d10, d01, d11);
    int c0 = fp * 32 + (lane & 15), c1 = c0 + 16;
    float bb0 = b_right[c0], bb1 = b_right[c1];
    for (int r = 0; r < 8; ++r) {
      int row = r + ((lane >> 4) << 3);
      if (n0r + row < NNODE) {
        rgt[(size_t)(n0r + row) * 128 + c0] = d00[r] + bb0;
        rgt[(size_t)(n0r + row) * 128 + c1] = d01[r] + bb1;
      }
      if (n0r + row + 16 < NNODE) {
        rgt[(size_t)(n0r + row + 16) * 128 + c0] = d10[r] + bb0;
        rgt[(size_t)(n0r + row + 16) * 128 + c1] = d11[r] + bb1;
      }
    }
  }

  // atp = X @ w_tp0
  for (int fp = 0; fp < 4; ++fp) {
    v8f d00, d10, d01, d11;
    gemm_22(Xs, 132, sw_tp0, 4, 32, fp, lane, d00, d10, d01, d11);
    int c0 = fp * 32 + (lane & 15), c1 = c0 + 16;
    for (int r = 0; r < 8; ++r) {
      int row = r + ((lane >> 4) << 3);
      if (n0r + row < NNODE) {
        atp[(size_t)(n0r + row) * 128 + c0] = d00[r];
        atp[(size_t)(n0r + row) * 128 + c1] = d01[r];
      }
      if (n0r + row + 16 < NNODE) {
        atp[(size_t)(n0r + row + 16) * 128 + c0] = d10[r];
        atp[(size_t)(n0r + row + 16) * 128 + c1] = d11[r];
      }
    }
  }

  // pv[d] = Xv(32x64) @ w_tp1 (reuse Bs, stride 68)
  for (int d = 0; d < 3; ++d) {
    lds_fence();
    for (int i = lane; i < 32 * 64; i += 32) {
      int r = i >> 6, j = i & 63;
      int rr = n0r + r; if (rr > NNODE - 1) rr = NNODE - 1;
      Bs[r * 68 + j] = h[(size_t)rr * 320 + 128 + 3 * j + d];
    }
    lds_fence();
    for (int fp = 0; fp < 2; ++fp) {
      v8f d00, d10, d01, d11;
      gemm_22(Bs, 68, sw_tp1, 2, 16, fp, lane, d00, d10, d01, d11);
      int c0 = fp * 32 + (lane & 15), c1 = c0 + 16;
      for (int r = 0; r < 8; ++r) {
        int row = r + ((lane >> 4) << 3);
        if (n0r + row < NNODE) {
          pv[(size_t)(n0r + row) * 192 + d * 64 + c0] = d00[r];
          pv[(size_t)(n0r + row) * 192 + d * 64 + c1] = d01[r];
        }
        if (n0r + row + 16 < NNODE) {
          pv[(size_t)(n0r + row + 16) * 192 + d * 64 + c0] = d10[r];
          pv[(size_t)(n0r + row + 16) * 192 + d * 64 + c1] = d11[r];
        }
      }
    }
  }
}

// ---------------------------------------------------------------------------
// Kernel 2: atom head (32 targets per wave).
// ---------------------------------------------------------------------------
__global__ void __launch_bounds__(32) atom_head(
    const float* __restrict__ hs, const int* __restrict__ tni,
    const float* __restrict__ sw_at1, const float* __restrict__ b_at1,
    const float* __restrict__ sw_at2, const float* __restrict__ b_at2,
    float* __restrict__ out)
{
  __shared__ float Xs[32 * 132];
  __shared__ float Ts[32 * 132];
  __shared__ int idx[32];
  const int lane = threadIdx.x;
  const int t0   = blockIdx.x * 32;

  int tt = t0 + lane; if (tt > NTGT - 1) tt = NTGT - 1;
  idx[lane] = tni[tt];
  lds_fence();
  for (int i = lane; i < 32 * 32; i += 32) {
    int r = i >> 5, c = (i & 31) << 2;
    *(float4*)(&Xs[r * 132 + c]) = *(const float4*)(hs + (size_t)idx[r] * 128 + c);
  }
  lds_fence();
  for (int fp = 0; fp < 4; ++fp) {
    v8f d00, d10, d01, d11;
    gemm_22(Xs, 132, sw_at1, 4, 32, fp, lane, d00, d10, d01, d11);
    int c0 = fp * 32 + (lane & 15), c1 = c0 + 16;
    float bb0 = b_at1[c0], bb1 = b_at1[c1];
    for (int r = 0; r < 8; ++r) {
      int row = r + ((lane >> 4) << 3);
      Ts[row * 132 + c0] = silu(d00[r] + bb0);
      Ts[(row + 16) * 132 + c0] = silu(d10[r] + bb0);
      Ts[row * 132 + c1] = silu(d01[r] + bb1);
      Ts[(row + 16) * 132 + c1] = silu(d11[r] + bb1);
    }
  }
  lds_fence();
  {
    v8f d00, d10, d01, d11;
    gemm_22(Ts, 132, sw_at2, 1, 32, 0, lane, d00, d10, d01, d11);
    int c0 = lane & 15, c1 = c0 + 16;
    float bb0 = b_at2[c0], bb1 = b_at2[c1];
    for (int r = 0; r < 8; ++r) {
      int row = r + ((lane >> 4) << 3);
      if (t0 + row < NTGT) {
        out[(size_t)(t0 + row) * 32 + c0] = d00[r] + bb0;
        out[(size_t)(t0 + row) * 32 + c1] = d01[r] + bb1;
      }
      if (t0 + row + 16 < NTGT) {
        out[(size_t)(t0 + row + 16) * 32 + c0] = d10[r] + bb0;
        out[(size_t)(t0 + row + 16) * 32 + c1] = d11[r] + bb1;
      }
    }
  }
}

// ---------------------------------------------------------------------------
// Kernel 3: edge MLP + layernorm + force scatter (32 edges per wave).
//   X = [e_scalar(64) | e_high_norms(32) | lft[row]*rgt[col](128)]  (K=224)
// ---------------------------------------------------------------------------
__global__ void __launch_bounds__(32) edge_force(
    const float* __restrict__ efin, const int* __restrict__ eidx,
    const float* __restrict__ lft, const float* __restrict__ rgt,
    const float* __restrict__ rt,
    const float* __restrict__ sw_el1, const float* __restrict__ b_el1,
    const float* __restrict__ ln_g, const float* __restrict__ ln_b,
    const float* __restrict__ W_el2, const float* __restrict__ b_el2,
    float* __restrict__ delta)
{
  __shared__ float Xs[32 * 228];
  __shared__ float XL[32 * 132];
  __shared__ int ridx[32], cidx[32];
  const int lane = threadIdx.x;
  const int e0   = blockIdx.x * 32;

  // prefetch next tile's streaming e_final rows (gfx1250 global_prefetch)
  __builtin_prefetch(efin + (size_t)(e0 + 32) * 160, 0, 1);

  ridx[lane] = eidx[e0 + lane];
  cidx[lane] = eidx[NEDGE + e0 + lane];
  lds_fence();
  for (int i = lane; i < 32 * 16; i += 32) {           // e_scalar -> 0..63
    int r = i >> 4, c = (i & 15) << 2;
    *(float4*)(&Xs[r * 228 + c]) = *(const float4*)(efin + (size_t)(e0 + r) * 160 + c);
  }
  for (int i = lane; i < 32 * 32; i += 32) {           // norms -> 64..95
    int r = i >> 5, j = i & 31;
    const float* p = efin + (size_t)(e0 + r) * 160 + 64 + 3 * j;
    float x = p[0], y = p[1], z = p[2];
    Xs[r * 228 + 64 + j] = sqrtf(x * x + y * y + z * z);
  }
  for (int i = lane; i < 32 * 32; i += 32) {           // inter -> 96..223
    int r = i >> 5, c = (i & 31) << 2;
    float4 a = *(const float4*)(lft + (size_t)ridx[r] * 128 + c);
    float4 b = *(const float4*)(rgt + (size_t)cidx[r] * 128 + c);
    float4 o;
    o.x = a.x * b.x; o.y = a.y * b.y; o.z = a.z * b.z; o.w = a.w * b.w;
    *(float4*)(&Xs[r * 228 + 96 + c]) = o;
  }
  lds_fence();
  for (int fp = 0; fp < 4; ++fp) {
    v8f d00, d10, d01, d11;
    gemm_22(Xs, 228, sw_el1, 4, 56, fp, lane, d00, d10, d01, d11);
    int c0 = fp * 32 + (lane & 15), c1 = c0 + 16;
    float bb0 = b_el1[c0], bb1 = b_el1[c1];
    for (int r = 0; r < 8; ++r) {
      int row = r + ((lane >> 4) << 3);
      XL[row * 132 + c0] = d00[r] + bb0;
      XL[(row + 16) * 132 + c0] = d10[r] + bb0;
      XL[row * 132 + c1] = d01[r] + bb1;
      XL[(row + 16) * 132 + c1] = d11[r] + bb1;
    }
  }
  lds_fence();
  {
    const float* x = &XL[lane * 132];
    float s = 0.f, s2 = 0.f;
    for (int k = 0; k < 128; ++k) { float v = x[k]; s += v; s2 += v * v; }
    float mu   = s * (1.f / 128.f);
    float var  = s2 * (1.f / 128.f) - mu * mu;
    float rsig = rsqrtf(var + 1e-5f);
    float w = 0.f;
    for (int k = 0; k < 128; ++k) {
      float v = (x[k] - mu) * rsig * ln_g[k] + ln_b[k];
      w += silu(v) * W_el2[k];
    }
    w += b_el2[0];
    int re = ridx[lane], ce = cidx[lane];
    float rx = rt[re * 3 + 0] - rt[ce * 3 + 0];
    float ry = rt[re * 3 + 1] - rt[ce * 3 + 1];
    float rz = rt[re * 3 + 2] - rt[ce * 3 + 2];
    float dist = sqrtf(rx * rx + ry * ry + rz * rz) + 1e-8f;
    float sc = w / (dist * (dist + 1.f));
    atomicAdd(&delta[re * 3 + 0], sc * rx);
    atomicAdd(&delta[re * 3 + 1], sc * ry);
    atomicAdd(&delta[re * 3 + 2], sc * rz);
  }
}

// ---------------------------------------------------------------------------
// Kernel 4: predicted_r0 = (r_t + delta)[target_node_idx]
// ---------------------------------------------------------------------------
__global__ void finalize_r0(const float* __restrict__ rt,
                            const float* __restrict__ delta,
                            const int* __restrict__ tni,
                            float* __restrict__ out) {
  int i = blockIdx.x * blockDim.x + threadIdx.x;
  if (i < NTGT) {
    int t = tni[i];
    out[i * 3 + 0] = rt[t * 3 + 0] + delta[t * 3 + 0];
    out[i * 3 + 1] = rt[t * 3 + 1] + delta[t * 3 + 1];
    out[i * 3 + 2] = rt[t * 3 + 2] + delta[t * 3 + 2];
  }
}

// ---------------------------------------------------------------------------
// Kernel 5: bond head (32 target edges per wave). bi = [tp | e_final],
// K = 161 zero-padded to 164 (swizzle + LDS tile), N of W_bh2 padded to 32.
// ---------------------------------------------------------------------------
__global__ void __launch_bounds__(32) bond_head(
    const float* __restrict__ h, const float* __restrict__ efin,
    const int* __restrict__ eidx, const int* __restrict__ tei,
    const float* __restrict__ atp, const float* __restrict__ pv,
    const float* __restrict__ sw_bh1, const float* __restrict__ b_bh1,
    const float* __restrict__ sw_bh2, const float* __restrict__ b_bh2,
    float* __restrict__ out)
{
  __shared__ float Xs[32 * 164];
  __shared__ float Ts[32 * 132];
  const int lane = threadIdx.x;
  const int t0   = blockIdx.x * 32;

  {
    int te = tei[t0 + lane];
    int r2 = eidx[te], c2 = eidx[NEDGE + te];
    const float* ar = atp + (size_t)r2 * 128;
    const float* hc = h + (size_t)c2 * 320;
    float ds = 0.f;
    for (int k = 0; k < 128; ++k) ds += ar[k] * hc[k];
    const float* pr = pv + (size_t)r2 * 192;
    float dv = 0.f;
    for (int d = 0; d < 3; ++d)
      for (int j = 0; j < 64; ++j)
        dv += pr[d * 64 + j] * hc[128 + 3 * j + d];
    float tp = ds * (1.f / 128.f) + dv * (1.f / (64.f * 1.7320508075688772f));
    float* xr = &Xs[lane * 164];
    xr[0] = tp;
    const float* er = efin + (size_t)te * 160;
    for (int j = 0; j < 160; ++j) xr[1 + j] = er[j];
    xr[161] = 0.f; xr[162] = 0.f; xr[163] = 0.f;
  }
  lds_fence();
  for (int fp = 0; fp < 4; ++fp) {
    v8f d00, d10, d01, d11;
    gemm_22(Xs, 164, sw_bh1, 4, 41, fp, lane, d00, d10, d01, d11);
    int c0 = fp * 32 + (lane & 15), c1 = c0 + 16;
    float bb0 = b_bh1[c0], bb1 = b_bh1[c1];
    for (int r = 0; r < 8; ++r) {
      int row = r + ((lane >> 4) << 3);
      Ts[row * 132 + c0] = silu(d00[r] + bb0);
      Ts[(row + 16) * 132 + c0] = silu(d10[r] + bb0);
      Ts[row * 132 + c1] = silu(d01[r] + bb1);
      Ts[(row + 16) * 132 + c1] = silu(d11[r] + bb1);
    }
  }
  lds_fence();
  {
    v8f d00, d10, d01, d11;
    gemm_22(Ts, 132, sw_bh2, 1, 32, 0, lane, d00, d10, d01, d11);
    int c0 = lane & 15;                 // cols 16..31 are zero padding
    if (c0 < 8) {
      float bb = b_bh2[c0];
      for (int r = 0; r < 8; ++r) {
        int row = r + ((lane >> 4) << 3);
        out[(size_t)(t0 + row) * 8 + c0] = d00[r] + bb;
        out[(size_t)(t0 + row + 16) * 8 + c0] = d10[r] + bb;
      }
    }
  }
}

// ---------------------------------------------------------------------------
extern "C" void kernel_launch(void* const* d_in, const int* in_sizes, int n_in,
                              void* d_out, int out_size, void* d_ws, size_t ws_size,
                              hipStream_t stream) {
  const float* h     = (const float*)d_in[0];
  const float* efin  = (const float*)d_in[1];
  const float* rt    = (const float*)d_in[2];
  const int*   tni   = (const int*)d_in[3];
  const int*   tei   = (const int*)d_in[4];
  const int*   eidx  = (const int*)d_in[5];
  const float* W_nsp = (const float*)d_in[6];
  const float* b_nsp = (const float*)d_in[7];
  const float* W_at1 = (const float*)d_in[8];
  const float* b_at1 = (const float*)d_in[9];
  const float* W_at2 = (const float*)d_in[10];
  const float* b_at2 = (const float*)d_in[11];
  const float* W_left  = (const float*)d_in[12];
  const float* b_left  = (const float*)d_in[13];
  const float* W_right = (const float*)d_in[14];
  const float* b_right = (const float*)d_in[15];
  const float* W_el1 = (const float*)d_in[16];
  const float* b_el1 = (const float*)d_in[17];
  const float* ln_g  = (const float*)d_in[18];
  const float* ln_b  = (const float*)d_in[19];
  const float* W_el2 = (const float*)d_in[20];
  const float* b_el2 = (const float*)d_in[21];
  const float* w_tp0 = (const float*)d_in[22];
  const float* w_tp1 = (const float*)d_in[23];
  const float* W_bh1 = (const float*)d_in[24];
  const float* b_bh1 = (const float*)d_in[25];
  const float* W_bh2 = (const float*)d_in[26];
  const float* b_bh2 = (const float*)d_in[27];

  float* out      = (float*)d_out;
  float* out_atom = out;            // 10000 x 32
  float* out_r0   = out + 320000;   // 10000 x 3
  float* out_bond = out + 350000;   // 100000 x 8

  float* ws    = (float*)d_ws;
  float* hs    = ws;
  float* lft   = ws + 6400000;
  float* rgt   = ws + 12800000;
  float* atp   = ws + 19200000;
  float* pvv   = ws + 25600000;     // N x 192
  float* delta = ws + 35200000;     // N x 3
  float* swz   = ws + 35350000;

  float* sw_nsp  = swz;               // 128x128 -> 16384
  float* sw_left = sw_nsp  + 16384;
  float* sw_right= sw_left + 16384;
  float* sw_tp0  = sw_right+ 16384;
  float* sw_at1  = sw_tp0  + 16384;
  float* sw_at2  = sw_at1  + 16384;   // 128x32  -> 4096
  float* sw_el1  = sw_at2  + 4096;    // 224x128 -> 28672
  float* sw_tp1  = sw_el1  + 28672;   // 64x64   -> 4096
  float* sw_bh1  = sw_tp1  + 4096;    // 164x128 -> 20992
  float* sw_bh2  = sw_bh1  + 20992;   // 128x32  -> 4096

  auto swzl = [&](const float* W, int Kw, int Nw, int Kp, int Np, float* o) {
    int total = (Kp / 4) * (Np / 32) * 32;
    swizzle_w<<<(total + 255) / 256, 256, 0, stream>>>(W, Kw, Nw, Kp, Np, o);
  };
  swzl(W_nsp,  128, 128, 128, 128, sw_nsp);
  swzl(W_left, 128, 128, 128, 128, sw_left);
  swzl(W_right,128, 128, 128, 128, sw_right);
  swzl(w_tp0,  128, 128, 128, 128, sw_tp0);
  swzl(W_at1,  128, 128, 128, 128, sw_at1);
  swzl(W_at2,  128,  32, 128,  32, sw_at2);
  swzl(W_el1,  224, 128, 224, 128, sw_el1);
  swzl(w_tp1,   64,  64,  64,  64, sw_tp1);
  swzl(W_bh1,  161, 128, 164, 128, sw_bh1);
  swzl(W_bh2,  128,   8, 128,  32, sw_bh2);

  zero_f32<<<(NNODE * 3 + 255) / 256, 256, 0, stream>>>(delta, NNODE * 3);

  node_precompute<<<(NNODE + 31) / 32, 32, 0, stream>>>(
      h, sw_nsp, b_nsp, sw_left, b_left, sw_right, b_right, sw_tp0, sw_tp1,
      hs, lft, rgt, atp, pvv);

  atom_head<<<(NTGT + 31) / 32, 32, 0, stream>>>(
      hs, tni, sw_at1, b_at1, sw_at2, b_at2, out_atom);

  edge_force<<<NEDGE / 32, 32, 0, stream>>>(
      efin, eidx, lft, rgt, rt, sw_el1, b_el1, ln_g, ln_b, W_el2, b_el2, delta);

  finalize_r0<<<(NTGT + 255) / 256, 256, 0, stream>>>(rt, delta, tni, out_r0);

  bond_head<<<ETGT / 32, 32, 0, stream>>>(
      h, efin, eidx, tei, atp, pvv, sw_bh1, b_bh1, sw_bh2, b_bh2, out_bond);
}